// FOCUS_54623394071212
// MI455X (gfx1250) — compile-verified
//
#include <hip/hip_runtime.h>

typedef float v2f __attribute__((ext_vector_type(2)));
typedef float v8f __attribute__((ext_vector_type(8)));

#define DC_B   32
#define DC_H   10
#define DC_W   32
#define DC_HW  320      // H*W
#define DC_G   16
#define DC_CG  32
#define DC_KK  9
#define DC_C   512
#define DC_O   512
#define DC_M   10240    // B*H*W
#define BN_T   32
#define BN_N   10240    // samples per channel = 320*32

// ---------------------------------------------------------------------------
// Kernel 1: per-channel BN statistics -> scale/shift
// ---------------------------------------------------------------------------
__global__ __launch_bounds__(256) void bn_stats_kernel(
    const float* __restrict__ x, const float* __restrict__ gamma,
    const float* __restrict__ beta, float* __restrict__ scale,
    float* __restrict__ shift)
{
    __shared__ float s1[256];
    __shared__ float s2[256];
    const int c   = blockIdx.x;
    const int tid = threadIdx.x;
    float sum = 0.f, sq = 0.f;
    for (int e = tid; e < BN_N; e += 256) {
        const int bb = e >> 5;
        const int t  = e & 31;
        const float v = x[((size_t)bb * DC_C + c) * BN_T + t];
        sum += v; sq += v * v;
    }
    s1[tid] = sum; s2[tid] = sq;
    __syncthreads();
    for (int s = 128; s > 0; s >>= 1) {
        if (tid < s) { s1[tid] += s1[tid + s]; s2[tid] += s2[tid + s]; }
        __syncthreads();
    }
    if (tid == 0) {
        const float mean = s1[0] * (1.f / BN_N);
        const float var  = s2[0] * (1.f / BN_N) - mean * mean;
        const float inv  = 1.f / sqrtf(var + 1e-5f);
        const float sc   = gamma[c] * inv;
        scale[c] = sc;
        shift[c] = beta[c] - mean * sc;
    }
}

// ---------------------------------------------------------------------------
// Kernel 2: normalize + relayout x -> Xbuf[b][g][h][w][c]  (Cg contiguous)
// ---------------------------------------------------------------------------
__global__ __launch_bounds__(256) void bn_apply_layout_kernel(
    const float* __restrict__ x, const float* __restrict__ scale,
    const float* __restrict__ shift, float* __restrict__ Xbuf)
{
    const int e = blockIdx.x * 256 + threadIdx.x;   // grid sized exactly
    int idx = e;
    const int c = idx & 31;  idx >>= 5;
    const int w = idx & 31;  idx >>= 5;
    const int h = idx % 10;  idx /= 10;
    const int g = idx & 15;  idx >>= 4;
    const int b = idx;                               // 0..31
    const int ch = (g << 5) | c;
    const float v = x[(((size_t)(b * 10 + h)) * DC_C + ch) * DC_W + w];
    Xbuf[e] = scale[ch] * v + shift[ch];
}

// ---------------------------------------------------------------------------
// Kernel 3: weight (O, C, KK) -> K-pair-interleaved B matrix
//   WtP[kpair][o][p]  with krow = kpair*2 + p,  krow = (g*KK + k)*Cg + c
//   => one global_load_b64 yields (row kk, row kk+1) at column o.
// ---------------------------------------------------------------------------
__global__ __launch_bounds__(256) void wt_transpose_kernel(
    const float* __restrict__ wgt, float* __restrict__ WtP)
{
    const int e = blockIdx.x * 256 + threadIdx.x;   // 4608*512 elements
    const int p     = e & 1;
    const int o     = (e >> 1) & 511;
    const int kpair = e >> 10;
    const int krow  = (kpair << 1) | p;
    const int c  = krow & 31;
    const int gk = krow >> 5;        // g*9 + k
    const int k  = gk % 9;
    const int g  = gk / 9;
    WtP[e] = wgt[((size_t)o * DC_C + ((g << 5) | c)) * DC_KK + k];
}

// ---------------------------------------------------------------------------
// Bilinear corner gather (4 contiguous channels), out-of-bounds -> 0
// ---------------------------------------------------------------------------
__device__ __forceinline__ void corner_fma(const float* __restrict__ Xbg,
                                           int y, int x, int cq, float wgt,
                                           float4& s)
{
    const bool inb = ((unsigned)y < (unsigned)DC_H) & ((unsigned)x < (unsigned)DC_W);
    const int yc = y < 0 ? 0 : (y > DC_H - 1 ? DC_H - 1 : y);
    const int xc = x < 0 ? 0 : (x > DC_W - 1 ? DC_W - 1 : x);
    const float4 v = *(const float4*)(Xbg + (size_t)((yc << 5) + xc) * DC_CG + cq);
    const float f = inb ? wgt : 0.f;
    s.x += f * v.x; s.y += f * v.y; s.z += f * v.z; s.w += f * v.w;
}

// ---------------------------------------------------------------------------
// Kernel 4: fused deformable-sample + fp32 WMMA GEMM + bias + output layout.
//   Block: 32 M-rows, 256 threads (8 waves), wave owns 64 N-columns.
//   Per (g,k) tap: build 32x32 A-slice in LDS, 8 K-steps of WMMA 16x16x4 f32.
// ---------------------------------------------------------------------------
__global__ __launch_bounds__(256) void dcn_gemm_kernel(
    const float* __restrict__ Xbuf, const float* __restrict__ offs,
    const float* __restrict__ mask, const float* __restrict__ WtP,
    const float* __restrict__ bias, float* __restrict__ out)
{
    __shared__ float As[32][36];   // 36-stride: bank-conflict-free, float4-aligned

    const int tid    = threadIdx.x;
    const int lane   = tid & 31;
    const int waveId = tid >> 5;
    const int hi     = lane >> 4;      // 0: lanes 0-15, 1: lanes 16-31
    const int laneM  = lane & 15;
    const int waveN  = waveId * 64;
    const int blockM = blockIdx.x * 32;

    // build-phase mapping: thread -> (row r in tile, 4-channel quad cq)
    const int r   = tid >> 3;          // 0..31
    const int cq  = (tid & 7) << 2;    // 0,4,...,28
    const int mg  = blockM + r;
    const int bb  = mg / DC_HW;
    const int rem = mg - bb * DC_HW;   // h*32 + w
    const int hh  = rem >> 5;
    const int ww  = rem & 31;

    v8f acc[2][4];
#pragma unroll
    for (int ms = 0; ms < 2; ++ms)
#pragma unroll
        for (int nt = 0; nt < 4; ++nt)
            acc[ms][nt] = (v8f)0.f;

    for (int g = 0; g < DC_G; ++g) {
        const float* Xbg = Xbuf + (size_t)((bb * DC_G + g) * DC_HW) * DC_CG;
#pragma unroll 1
        for (int k = 0; k < DC_KK; ++k) {
            const int tap = g * DC_KK + k;
            // ---- build A slice: As[r][c] = mask * bilinear(X[b,g,:,py,px])[c]
            {
                const float dy = offs[((size_t)bb * 288 + tap * 2    ) * DC_HW + rem];
                const float dx = offs[((size_t)bb * 288 + tap * 2 + 1) * DC_HW + rem];
                const float mv = mask[((size_t)bb * 144 + tap        ) * DC_HW + rem];
                const float py = dy + (float)(k / 3 + hh - 1);
                const float px = dx + (float)(k % 3 + ww - 1);
                const float y0f = floorf(py), x0f = floorf(px);
                const float ly = py - y0f, lx = px - x0f;
                const int y0 = (int)y0f, x0 = (int)x0f;
                const float w00 = (1.f - ly) * (1.f - lx) * mv;
                const float w01 = (1.f - ly) * lx * mv;
                const float w10 = ly * (1.f - lx) * mv;
                const float w11 = ly * lx * mv;
                float4 s = {0.f, 0.f, 0.f, 0.f};
                corner_fma(Xbg, y0,     x0,     cq, w00, s);
                corner_fma(Xbg, y0,     x0 + 1, cq, w01, s);
                corner_fma(Xbg, y0 + 1, x0,     cq, w10, s);
                corner_fma(Xbg, y0 + 1, x0 + 1, cq, w11, s);
                *(float4*)&As[r][cq] = s;
            }
            __syncthreads();

            // ---- WMMA over the 32-wide K slice of this tap
            const int gkBase = tap * DC_CG;
#pragma unroll
            for (int k0 = 0; k0 < DC_CG; k0 += 4) {
                // A fragment (16x4 f32): VGPR0 = K(2*hi), VGPR1 = K(2*hi+1)
                const int kk = k0 + 2 * hi;
                const v2f a0 = *(const v2f*)&As[laneM][kk];
                const v2f a1 = *(const v2f*)&As[16 + laneM][kk];
                // B fragment: one b64 load per N-subtile from pair-interleaved WtP
                const float* Bp =
                    WtP + ((size_t)(((gkBase + kk) >> 1)) * DC_O + waveN + laneM) * 2;
#pragma unroll
                for (int nt = 0; nt < 4; ++nt) {
                    const v2f bf = *(const v2f*)(Bp + nt * 32);  // rows kk, kk+1
                    acc[0][nt] = __builtin_amdgcn_wmma_f32_16x16x4_f32(
                        false, a0, false, bf, (short)0, acc[0][nt], false, false);
                    acc[1][nt] = __builtin_amdgcn_wmma_f32_16x16x4_f32(
                        false, a1, false, bf, (short)0, acc[1][nt], false, false);
                }
            }
            __syncthreads();
        }
    }

    // ---- epilogue: C/D layout -> out[(b*10+h), o, w] = acc + bias[o]
#pragma unroll
    for (int ms = 0; ms < 2; ++ms) {
#pragma unroll
        for (int nt = 0; nt < 4; ++nt) {
            const int o  = waveN + nt * 16 + laneM;
            const float bo = bias[o];
#pragma unroll
            for (int i = 0; i < 8; ++i) {
                const int mgl = blockM + ms * 16 + (hi << 3) + i;
                out[((size_t)(mgl >> 5) * DC_O + o) * DC_W + (mgl & 31)] =
                    acc[ms][nt][i] + bo;
            }
        }
    }
}

// ---------------------------------------------------------------------------
// Host launcher
// ---------------------------------------------------------------------------
extern "C" void kernel_launch(void* const* d_in, const int* in_sizes, int n_in,
                              void* d_out, int out_size, void* d_ws, size_t ws_size,
                              hipStream_t stream)
{
    const float* x      = (const float*)d_in[0];   // (320,512,32)
    const float* offset = (const float*)d_in[1];   // (32,288,10,32)
    const float* mask   = (const float*)d_in[2];   // (32,144,10,32)
    const float* gamma  = (const float*)d_in[3];   // (512,)
    const float* beta   = (const float*)d_in[4];   // (512,)
    const float* weight = (const float*)d_in[5];   // (512,512,3,3)
    const float* bias   = (const float*)d_in[6];   // (512,)
    float* out = (float*)d_out;                    // (320,512,32)

    // workspace layout
    float* scale = (float*)d_ws;                               // 512
    float* shift = scale + 512;                                // 512
    float* Xbuf  = (float*)((char*)d_ws + 4096);               // 5,242,880 f32
    float* WtP   = Xbuf + 5242880;                             // 2,359,296 f32
    (void)in_sizes; (void)n_in; (void)out_size; (void)ws_size;

    bn_stats_kernel<<<DC_C, 256, 0, stream>>>(x, gamma, beta, scale, shift);
    bn_apply_layout_kernel<<<5242880 / 256, 256, 0, stream>>>(x, scale, shift, Xbuf);
    wt_transpose_kernel<<<2359296 / 256, 256, 0, stream>>>(weight, WtP);
    dcn_gemm_kernel<<<DC_M / 32, 256, 0, stream>>>(Xbuf, offset, mask, WtP, bias, out);
}